// NMS_58007828300125
// MI455X (gfx1250) — compile-verified
//
#include <hip/hip_runtime.h>
#include <stdint.h>

// Batched greedy NMS for MI455X (gfx1250, wave32).
// One workgroup per batch: scores (256KB) staged into LDS via CDNA5
// global_load_async_to_lds_b128 (ASYNCcnt), then k=6 fused
// suppress+argmax sweeps entirely out of LDS.

#define NBOX    65536
#define KPROP   6
#define NTHREADS 1024                      // 32 wave32 waves
#define CHUNK4  (NBOX / 4 / NTHREADS)      // 16 float4 per thread
#define IOU_THR 0.25f
#define NEG_INF (-__builtin_inff())

__global__ __launch_bounds__(NTHREADS, 1)
void nms_kernel(const float* __restrict__ scores,
                const float* __restrict__ anchors,
                int* __restrict__ out)
{
    __shared__ __align__(16) float s_scores[NBOX];   // 256 KB (of 320 KB/WGP)
    __shared__ float s_wv[32];
    __shared__ int   s_wi[32];
    __shared__ float s_box[5];   // y1, x1, y2, x2, area of selected box
    __shared__ int   s_selidx;

    const int tid = threadIdx.x;
    const int b   = blockIdx.x;
    const float* bs = scores + (size_t)b * (size_t)NBOX;

    // ---------------- async copy: global scores -> LDS ----------------
    // GVS mode: saddr = batch base (SGPR pair), vaddr = per-lane byte
    // offset, dst VGPR = per-lane LDS byte address. Each instruction
    // moves 16B/lane -> 16 issues cover the whole 256KB block-wide.
    {
        #pragma unroll
        for (int i = 0; i < CHUNK4; ++i) {
            const int e4 = tid + i * NTHREADS;                 // float4 index
            const uint32_t goff = (uint32_t)e4 * 16u;          // global byte offset
            const uint32_t lds  = (uint32_t)(uintptr_t)(&s_scores[e4 * 4]);
            asm volatile("global_load_async_to_lds_b128 %0, %1, %2"
                         :: "v"(lds), "v"(goff), "s"(bs)
                         : "memory");
        }
        // Wait for this wave's async transfers (ASYNCcnt) and for address
        // translation retirement (XCNT; NOP when XNACK replay is off) before
        // any VGPR that fed an async address can be considered dead.
        asm volatile("s_wait_asynccnt 0x0\n\t"
                     "s_wait_xcnt 0x0" ::: "memory");
    }
    __syncthreads();

    const float4* s4  = (const float4*)s_scores;
    float4*       s4w = (float4*)s_scores;
    const float4* a4  = (const float4*)anchors;

    float sy1 = 0.f, sx1 = 0.f, sy2 = 0.f, sx2 = 0.f, sarea = 0.f;

    for (int p = 0; p < KPROP; ++p) {
        float best = NEG_INF;
        int   bidx = 0x7FFFFFFF;

        if (p == 0) {
            // plain argmax sweep
            #pragma unroll
            for (int i = 0; i < CHUNK4; ++i) {
                const int q = tid + i * NTHREADS;
                const float4 sv = s4[q];
                const int e = q * 4;
                float vs[4] = {sv.x, sv.y, sv.z, sv.w};
                #pragma unroll
                for (int c = 0; c < 4; ++c) {
                    const float v = vs[c];
                    const int   ix = e + c;
                    if (v > best || (v == best && ix < bidx)) { best = v; bidx = ix; }
                }
            }
        } else {
            // fused: suppress by previously selected box, then argmax
            #pragma unroll 4
            for (int i = 0; i < CHUNK4; ++i) {
                const int q = tid + i * NTHREADS;
                float4 sv = s4[q];
                const int e = q * 4;
                float vs[4] = {sv.x, sv.y, sv.z, sv.w};
                #pragma unroll
                for (int c = 0; c < 4; ++c) {
                    const float4 bx = a4[e + c];           // y1,x1,y2,x2 (L2-resident)
                    const float iy1 = fmaxf(sy1, bx.x);
                    const float ix1 = fmaxf(sx1, bx.y);
                    const float iy2 = fminf(sy2, bx.z);
                    const float ix2 = fminf(sx2, bx.w);
                    const float inter = fmaxf(iy2 - iy1, 0.f) * fmaxf(ix2 - ix1, 0.f);
                    const float areaj = (bx.z - bx.x) * (bx.w - bx.y);
                    const float uni = sarea + areaj - inter;
                    const float iou = inter / uni;
                    if (iou > IOU_THR) vs[c] = NEG_INF;    // selected box self-suppresses (iou=1)
                    const float v = vs[c];
                    const int   ix = e + c;
                    if (v > best || (v == best && ix < bidx)) { best = v; bidx = ix; }
                }
                sv.x = vs[0]; sv.y = vs[1]; sv.z = vs[2]; sv.w = vs[3];
                s4w[q] = sv;                                // persist -inf marks
            }
        }

        // ---------------- block argmax (wave32 shuffle + 32-entry LDS) --------
        #pragma unroll
        for (int off = 16; off > 0; off >>= 1) {
            const float ov = __shfl_down(best, off, 32);
            const int   oi = __shfl_down(bidx, off, 32);
            if (ov > best || (ov == best && oi < bidx)) { best = ov; bidx = oi; }
        }
        const int wave = tid >> 5;
        const int lane = tid & 31;
        if (lane == 0) { s_wv[wave] = best; s_wi[wave] = bidx; }
        __syncthreads();
        if (tid < 32) {
            float v = s_wv[tid];
            int   ix = s_wi[tid];
            #pragma unroll
            for (int off = 16; off > 0; off >>= 1) {
                const float ov = __shfl_down(v, off, 32);
                const int   oi = __shfl_down(ix, off, 32);
                if (ov > v || (ov == v && oi < ix)) { v = ov; ix = oi; }
            }
            if (tid == 0) {
                s_selidx = ix;
                out[b * KPROP + p] = ix;
                const float4 bx = a4[ix];
                s_box[0] = bx.x; s_box[1] = bx.y;
                s_box[2] = bx.z; s_box[3] = bx.w;
                s_box[4] = (bx.z - bx.x) * (bx.w - bx.y);
            }
        }
        __syncthreads();
        sy1 = s_box[0]; sx1 = s_box[1]; sy2 = s_box[2]; sx2 = s_box[3];
        sarea = s_box[4];
        (void)s_selidx;
    }
}

extern "C" void kernel_launch(void* const* d_in, const int* in_sizes, int n_in,
                              void* d_out, int out_size, void* d_ws, size_t ws_size,
                              hipStream_t stream) {
    (void)n_in; (void)out_size; (void)d_ws; (void)ws_size;
    const float* scores  = (const float*)d_in[0];   // (B, N) f32
    const float* anchors = (const float*)d_in[1];   // (N, 4) f32
    int* out = (int*)d_out;                          // (B, K) int32
    const int B = in_sizes[0] / NBOX;                // 512
    nms_kernel<<<dim3(B), dim3(NTHREADS), 0, stream>>>(scores, anchors, out);
}